// PillarFeatureNet_47433618817536
// MI455X (gfx1250) — compile-verified
//
#include <hip/hip_runtime.h>

// ---- problem constants (fixed by the reference) ----
#define P_CNT   40000
#define NPTS    32
#define C_AUG_P 12      // 9 real channels padded to 12 (K = 3 x 4)
#define C_OUT_N 64
#define VX      0.2f
#define VY      0.2f
#define X_OFF   0.1f    // VX/2 + 0.0
#define Y_OFF   (-39.9f) // VY/2 - 40.0
#define BN_EPS  1e-3f

typedef __attribute__((ext_vector_type(2))) float v2f;
typedef __attribute__((ext_vector_type(8))) float v8f;

// Build the 12 (padded) augmented channels for this lane's point and stage them
// into this wave's LDS slab. Returns nothing; LDS slab laid out [point][12].
__device__ __forceinline__ void build_feats_to_lds(
    const float* __restrict__ feat, const int* __restrict__ nvox,
    const int* __restrict__ coors, int p, int lane, float* __restrict__ myfeat)
{
    const float4 f = ((const float4*)feat)[p * NPTS + lane];
    const int   nv = nvox[p];
    const int   cy = coors[p * 4 + 2];
    const int   cx = coors[p * 4 + 3];

    // pillar xyz mean: sum over ALL 32 points (reference has no mask here), / nv
    float sx = f.x, sy = f.y, sz = f.z;
    #pragma unroll
    for (int off = 16; off >= 1; off >>= 1) {
        sx += __shfl_xor(sx, off, 32);
        sy += __shfl_xor(sy, off, 32);
        sz += __shfl_xor(sz, off, 32);
    }
    const float inv = 1.0f / (float)nv;
    const float mx = sx * inv, my = sy * inv, mz = sz * inv;

    const float mask = (lane < nv) ? 1.0f : 0.0f;
    const float fcx = f.x - ((float)cx * VX + X_OFF);
    const float fcy = f.y - ((float)cy * VY + Y_OFF);

    float4* dst = (float4*)(myfeat + lane * C_AUG_P);
    dst[0] = make_float4(f.x * mask, f.y * mask, f.z * mask, f.w * mask);
    dst[1] = make_float4((f.x - mx) * mask, (f.y - my) * mask,
                         (f.z - mz) * mask, fcx * mask);
    dst[2] = make_float4(fcy * mask, 0.0f, 0.0f, 0.0f);
    // LDS ops from the same wave are in-order in HW; just stop compiler reorder.
    asm volatile("" ::: "memory");
}

// Load B fragments once: B[k][col] = W[col*9 + k], zero-padded to K=12.
// v2f layout per (ntile, kstep): lanes0-15 hold K=4ks+{0,1}, lanes16-31 K=4ks+{2,3}.
__device__ __forceinline__ void load_b_frags(const float* __restrict__ W,
                                             int hl, int ll, v2f b[4][3])
{
    #pragma unroll
    for (int nt = 0; nt < 4; ++nt)
        #pragma unroll
        for (int ks = 0; ks < 3; ++ks)
            #pragma unroll
            for (int j = 0; j < 2; ++j) {
                const int k   = 4 * ks + 2 * hl + j;
                const int col = 16 * nt + ll;
                b[nt][ks][j] = (k < 9) ? W[col * 9 + k] : 0.0f;
            }
}

// A fragments for one 16-row M tile: lanes0-15 K=4ks+{0,1}, lanes16-31 K=4ks+{2,3}.
__device__ __forceinline__ void load_a_frags(const float* __restrict__ myfeat,
                                             int mt, int hl, int ll, v2f a[3])
{
    const float* src = myfeat + (16 * mt + ll) * C_AUG_P + 2 * hl;
    a[0][0] = src[0]; a[0][1] = src[1];
    a[1][0] = src[4]; a[1][1] = src[5];
    a[2][0] = src[8]; a[2][1] = src[9];
}

// -------- pass 1: accumulate per-channel sum(x) and sum(x^2) --------
__global__ __launch_bounds__(256) void pfn_stats(
    const float* __restrict__ feat, const int* __restrict__ nvox,
    const int* __restrict__ coors, const float* __restrict__ W,
    float* __restrict__ gsum, float* __restrict__ gsq)
{
    __shared__ float sfeat[8 * NPTS * C_AUG_P];
    __shared__ float wsum[C_OUT_N], wsq[C_OUT_N];

    const int tid = threadIdx.x;
    const int wave = tid >> 5, lane = tid & 31;
    const int hl = lane >> 4, ll = lane & 15;

    if (tid < C_OUT_N) { wsum[tid] = 0.0f; wsq[tid] = 0.0f; }
    __syncthreads();

    v2f b[4][3];
    load_b_frags(W, hl, ll, b);

    float sacc[4]  = {0.f, 0.f, 0.f, 0.f};
    float s2acc[4] = {0.f, 0.f, 0.f, 0.f};

    float* myfeat = sfeat + wave * (NPTS * C_AUG_P);
    const int nwaves = gridDim.x * 8;
    for (int p = blockIdx.x * 8 + wave; p < P_CNT; p += nwaves) {
        build_feats_to_lds(feat, nvox, coors, p, lane, myfeat);
        #pragma unroll
        for (int mt = 0; mt < 2; ++mt) {
            v2f a[3];
            load_a_frags(myfeat, mt, hl, ll, a);
            #pragma unroll
            for (int nt = 0; nt < 4; ++nt) {
                v8f d = {0.f, 0.f, 0.f, 0.f, 0.f, 0.f, 0.f, 0.f};
                d = __builtin_amdgcn_wmma_f32_16x16x4_f32(false, a[0], false, b[nt][0], (short)0, d, false, false);
                d = __builtin_amdgcn_wmma_f32_16x16x4_f32(false, a[1], false, b[nt][1], (short)0, d, false, false);
                d = __builtin_amdgcn_wmma_f32_16x16x4_f32(false, a[2], false, b[nt][2], (short)0, d, false, false);
                float s = 0.f, s2 = 0.f;
                #pragma unroll
                for (int r = 0; r < 8; ++r) { const float v = d[r]; s += v; s2 = fmaf(v, v, s2); }
                sacc[nt] += s; s2acc[nt] += s2;
            }
        }
    }

    // fold the two half-wave row groups (same channel in lane L and L^16)
    #pragma unroll
    for (int nt = 0; nt < 4; ++nt) {
        sacc[nt]  += __shfl_xor(sacc[nt], 16, 32);
        s2acc[nt] += __shfl_xor(s2acc[nt], 16, 32);
    }
    if (lane < 16) {
        #pragma unroll
        for (int nt = 0; nt < 4; ++nt) {
            atomicAdd(&wsum[16 * nt + lane], sacc[nt]);
            atomicAdd(&wsq[16 * nt + lane], s2acc[nt]);
        }
    }
    __syncthreads();
    if (tid < C_OUT_N) {
        atomicAdd(&gsum[tid], wsum[tid]);
        atomicAdd(&gsq[tid],  wsq[tid]);
    }
}

// -------- finalize: BN scale/bias per channel --------
__global__ void pfn_finalize(const float* __restrict__ gsum, const float* __restrict__ gsq,
                             const float* __restrict__ gamma, const float* __restrict__ beta,
                             float* __restrict__ scale, float* __restrict__ bias)
{
    const int c = threadIdx.x;
    if (c < C_OUT_N) {
        const float invn = 1.0f / (float)(P_CNT * NPTS);
        const float mean = gsum[c] * invn;
        const float var  = gsq[c] * invn - mean * mean;
        const float s    = gamma[c] * rsqrtf(var + BN_EPS);
        scale[c] = s;
        bias[c]  = beta[c] - mean * s;
    }
}

// -------- pass 2: recompute x, normalize, relu, max over N, write out --------
__global__ __launch_bounds__(256) void pfn_out(
    const float* __restrict__ feat, const int* __restrict__ nvox,
    const int* __restrict__ coors, const float* __restrict__ W,
    const float* __restrict__ scale, const float* __restrict__ bias,
    float* __restrict__ out)
{
    __shared__ float sfeat[8 * NPTS * C_AUG_P];
    const int tid = threadIdx.x;
    const int wave = tid >> 5, lane = tid & 31;
    const int hl = lane >> 4, ll = lane & 15;

    const int p = blockIdx.x * 8 + wave;
    if (p >= P_CNT) return;   // wave-uniform

    v2f b[4][3];
    load_b_frags(W, hl, ll, b);

    float sc[4], bi[4];
    #pragma unroll
    for (int nt = 0; nt < 4; ++nt) {
        sc[nt] = scale[16 * nt + ll];
        bi[nt] = bias[16 * nt + ll];
    }

    float* myfeat = sfeat + wave * (NPTS * C_AUG_P);
    build_feats_to_lds(feat, nvox, coors, p, lane, myfeat);

    float m[4] = {0.f, 0.f, 0.f, 0.f};   // relu() >= 0, so 0 is a safe identity
    #pragma unroll
    for (int mt = 0; mt < 2; ++mt) {
        v2f a[3];
        load_a_frags(myfeat, mt, hl, ll, a);
        #pragma unroll
        for (int nt = 0; nt < 4; ++nt) {
            v8f d = {0.f, 0.f, 0.f, 0.f, 0.f, 0.f, 0.f, 0.f};
            d = __builtin_amdgcn_wmma_f32_16x16x4_f32(false, a[0], false, b[nt][0], (short)0, d, false, false);
            d = __builtin_amdgcn_wmma_f32_16x16x4_f32(false, a[1], false, b[nt][1], (short)0, d, false, false);
            d = __builtin_amdgcn_wmma_f32_16x16x4_f32(false, a[2], false, b[nt][2], (short)0, d, false, false);
            #pragma unroll
            for (int r = 0; r < 8; ++r)
                m[nt] = fmaxf(m[nt], fmaxf(0.0f, fmaf(d[r], sc[nt], bi[nt])));
        }
    }
    #pragma unroll
    for (int nt = 0; nt < 4; ++nt)
        m[nt] = fmaxf(m[nt], __shfl_xor(m[nt], 16, 32));

    if (lane < 16) {
        #pragma unroll
        for (int nt = 0; nt < 4; ++nt)
            out[p * C_OUT_N + 16 * nt + lane] = m[nt];
    }
}

// -------- zero the stat accumulators (graph-replay safe) --------
__global__ void pfn_zero(float* __restrict__ ws)
{
    if (threadIdx.x < 128) ws[threadIdx.x] = 0.0f;
}

extern "C" void kernel_launch(void* const* d_in, const int* in_sizes, int n_in,
                              void* d_out, int out_size, void* d_ws, size_t ws_size,
                              hipStream_t stream) {
    const float* features   = (const float*)d_in[0];
    const int*   num_voxels = (const int*)d_in[1];
    const int*   coors      = (const int*)d_in[2];
    const float* linear_w   = (const float*)d_in[3];
    const float* gamma      = (const float*)d_in[4];
    const float* beta       = (const float*)d_in[5];
    float* ws    = (float*)d_ws;
    float* gsum  = ws;
    float* gsq   = ws + 64;
    float* scale = ws + 128;
    float* bias  = ws + 192;
    float* out   = (float*)d_out;

    pfn_zero<<<1, 128, 0, stream>>>(ws);
    pfn_stats<<<512, 256, 0, stream>>>(features, num_voxels, coors, linear_w, gsum, gsq);
    pfn_finalize<<<1, 64, 0, stream>>>(gsum, gsq, gamma, beta, scale, bias);
    pfn_out<<<(P_CNT + 7) / 8, 256, 0, stream>>>(features, num_voxels, coors, linear_w,
                                                 scale, bias, out);
}